// GCN_3281355014801
// MI455X (gfx1250) — compile-verified
//
#include <hip/hip_runtime.h>
#include <math.h>

typedef __attribute__((ext_vector_type(2))) float v2f;
typedef __attribute__((ext_vector_type(8))) float v8f;

// ---------------------------------------------------------------- degree/norm
__global__ void init_deg_kernel(float* __restrict__ deg, int N) {
  int i = blockIdx.x * blockDim.x + threadIdx.x;
  if (i < N) deg[i] = 1.0f;                       // self-loop
}

__global__ void count_deg_kernel(const long long* __restrict__ dst,
                                 float* __restrict__ deg, long long E) {
  long long e = (long long)blockIdx.x * blockDim.x + threadIdx.x;
  if (e < E) atomicAdd(&deg[dst[e]], 1.0f);
}

__global__ void dis_kernel(float* __restrict__ deg, int N) {
  int i = blockIdx.x * blockDim.x + threadIdx.x;
  if (i < N) deg[i] = rsqrtf(deg[i]);             // deg >= 1 always; in place
}

// ---------------------------------------------------------------- WMMA GEMMs
// xw1[N,64] = x[N,128] @ W1[128,64], fp32 via V_WMMA_F32_16X16X4_F32.
// Block = 128 threads (4 wave32) -> 16 rows x 64 cols per block, 1 tile/wave.
__global__ void gemm1_wmma_kernel(const float* __restrict__ X,
                                  const float* __restrict__ W,
                                  float* __restrict__ XW, int N) {
  int wave = threadIdx.x >> 5;        // n-tile 0..3
  int lane = threadIdx.x & 31;
  int half = lane >> 4;               // K-split across lane halves
  int l    = lane & 15;
  int m0 = blockIdx.x * 16;
  int n0 = wave * 16;
  int row = m0 + l; if (row >= N) row = N - 1;    // clamped rows never stored
  const float* xrow = X + (size_t)row * 128;
  v8f acc = {};
  #pragma unroll 4
  for (int k = 0; k < 128; k += 4) {
    v2f a, b;
    // A 16x4: lanes 0-15 hold K=k,k+1 ; lanes 16-31 hold K=k+2,k+3
    a.x = xrow[k + 2 * half];
    a.y = xrow[k + 2 * half + 1];
    // B 4x16: rows striped across lanes, K split across halves like A
    b.x = W[(size_t)(k + 2 * half) * 64 + n0 + l];
    b.y = W[(size_t)(k + 2 * half + 1) * 64 + n0 + l];
    acc = __builtin_amdgcn_wmma_f32_16x16x4_f32(
        false, a, false, b, (short)0, acc, false, false);
  }
  // C/D: VGPR j -> row m0+j (lanes 0-15) / m0+8+j (lanes 16-31), col n0+lane%16
  float* outp = XW + (size_t)(m0 + half * 8) * 64 + n0 + l;
  if (m0 + 16 <= N) {                 // uniform fast path: whole tile in range
    #pragma unroll
    for (int j = 0; j < 8; ++j) outp[(size_t)j * 64] = acc[j];
  } else {
    #pragma unroll
    for (int j = 0; j < 8; ++j)
      if (m0 + half * 8 + j < N) outp[(size_t)j * 64] = acc[j];
  }
}

// xw2[N,40] = relu(agg1[N,64]) @ W2[64,40]; ReLU fused into A-fragment load.
// Block = 96 threads (3 wave32) -> 16 rows x 48 cols (cols >=40 masked).
__global__ void gemm2_wmma_kernel(const float* __restrict__ H,
                                  const float* __restrict__ W,
                                  float* __restrict__ XW, int N) {
  int wave = threadIdx.x >> 5;        // n-tile 0..2
  int lane = threadIdx.x & 31;
  int half = lane >> 4;
  int l    = lane & 15;
  int m0 = blockIdx.x * 16;
  int n0 = wave * 16;
  int row = m0 + l; if (row >= N) row = N - 1;
  const float* hrow = H + (size_t)row * 64;
  int col = n0 + l;
  bool cok = (col < 40);
  v8f acc = {};
  #pragma unroll 4
  for (int k = 0; k < 64; k += 4) {
    v2f a, b;
    a.x = fmaxf(hrow[k + 2 * half], 0.0f);        // fused ReLU
    a.y = fmaxf(hrow[k + 2 * half + 1], 0.0f);
    b.x = cok ? W[(size_t)(k + 2 * half) * 40 + col] : 0.0f;
    b.y = cok ? W[(size_t)(k + 2 * half + 1) * 40 + col] : 0.0f;
    acc = __builtin_amdgcn_wmma_f32_16x16x4_f32(
        false, a, false, b, (short)0, acc, false, false);
  }
  float* outp = XW + (size_t)(m0 + half * 8) * 40 + col;
  if (cok) {
    if (m0 + 16 <= N) {               // uniform fast path
      #pragma unroll
      for (int j = 0; j < 8; ++j) outp[(size_t)j * 40] = acc[j];
    } else {
      #pragma unroll
      for (int j = 0; j < 8; ++j)
        if (m0 + half * 8 + j < N) outp[(size_t)j * 40] = acc[j];
    }
  }
}

// ---------------------------------------------------------------- aggregation
// agg = bias + self-loop term (no atomics needed).
__global__ void init_agg_kernel(const float* __restrict__ xw,
                                const float* __restrict__ dis,
                                const float* __restrict__ bias,
                                float* __restrict__ agg, int N, int C) {
  long long idx = (long long)blockIdx.x * blockDim.x + threadIdx.x;
  long long tot = (long long)N * C;
  if (idx >= tot) return;
  int i = (int)(idx / C);
  int c = (int)(idx % C);
  float di = dis[i];
  agg[idx] = bias[c] + xw[idx] * di * di;
}

// One wave per edge, 64 cols: lane L handles cols L and L+32 (coalesced).
__global__ void edge_agg64_kernel(const long long* __restrict__ src,
                                  const long long* __restrict__ dst,
                                  const float* __restrict__ xw,
                                  const float* __restrict__ dis,
                                  float* __restrict__ agg, long long E) {
  long long e = (long long)blockIdx.x * (blockDim.x >> 5) + (threadIdx.x >> 5);
  if (e >= E) return;
  int lane = threadIdx.x & 31;
  long long s = src[e], d = dst[e];
  float nrm = dis[s] * dis[d];
  const float* xs = xw + (size_t)s * 64;
  float* od = agg + (size_t)d * 64;
  atomicAdd(&od[lane],      xs[lane]      * nrm);
  atomicAdd(&od[lane + 32], xs[lane + 32] * nrm);
}

// One wave per edge, 40 cols.
__global__ void edge_agg40_kernel(const long long* __restrict__ src,
                                  const long long* __restrict__ dst,
                                  const float* __restrict__ xw,
                                  const float* __restrict__ dis,
                                  float* __restrict__ agg, long long E) {
  long long e = (long long)blockIdx.x * (blockDim.x >> 5) + (threadIdx.x >> 5);
  if (e >= E) return;
  int lane = threadIdx.x & 31;
  long long s = src[e], d = dst[e];
  float nrm = dis[s] * dis[d];
  const float* xs = xw + (size_t)s * 40;
  float* od = agg + (size_t)d * 40;
  atomicAdd(&od[lane], xs[lane] * nrm);
  if (lane < 8) atomicAdd(&od[32 + lane], xs[32 + lane] * nrm);
}

// ---------------------------------------------------------------- log_softmax
__device__ __forceinline__ float wave_max32(float v) {
  #pragma unroll
  for (int off = 16; off > 0; off >>= 1) v = fmaxf(v, __shfl_xor(v, off, 32));
  return v;
}
__device__ __forceinline__ float wave_sum32(float v) {
  #pragma unroll
  for (int off = 16; off > 0; off >>= 1) v += __shfl_xor(v, off, 32);
  return v;
}

__global__ void logsoftmax40_kernel(const float* __restrict__ agg,
                                    float* __restrict__ out, int N) {
  int row = blockIdx.x * (blockDim.x >> 5) + (threadIdx.x >> 5);
  if (row >= N) return;
  int lane = threadIdx.x & 31;
  const float* r = agg + (size_t)row * 40;
  float v0 = r[lane];
  float v1 = (lane < 8) ? r[32 + lane] : -INFINITY;
  float m  = wave_max32(fmaxf(v0, v1));
  float s  = wave_sum32(expf(v0 - m) + ((lane < 8) ? expf(v1 - m) : 0.0f));
  float lz = m + logf(s);
  float* o = out + (size_t)row * 40;
  o[lane] = v0 - lz;
  if (lane < 8) o[32 + lane] = v1 - lz;
}

// ---------------------------------------------------------------- launcher
extern "C" void kernel_launch(void* const* d_in, const int* in_sizes, int n_in,
                              void* d_out, int out_size, void* d_ws, size_t ws_size,
                              hipStream_t stream) {
  const float*     x   = (const float*)d_in[0];
  const long long* ei  = (const long long*)d_in[1];   // int64 [2, E]
  const float*     W1  = (const float*)d_in[2];
  const float*     b1  = (const float*)d_in[3];
  const float*     W2  = (const float*)d_in[4];
  const float*     b2  = (const float*)d_in[5];
  float*           out = (float*)d_out;

  int       N = in_sizes[0] / 128;
  long long E = (long long)in_sizes[1] / 2;
  const long long* src = ei;
  const long long* dst = ei + E;

  // Scratch layout (aliased across phases): dis | bufA(xw) | bufB(agg)
  float* ws   = (float*)d_ws;
  size_t Npad = (((size_t)N) + 63) & ~(size_t)63;
  float* dis  = ws;
  float* bufA = ws + Npad;                 // xw1 [N,64] -> later xw2 [N,40]
  float* bufB = bufA + (size_t)N * 64;     // agg1 [N,64] -> later agg2 [N,40]

  int mtiles = (N + 15) / 16;
  long long eb8   = (E + 7) / 8;           // wave-per-edge, 8 waves/block
  long long eb256 = (E + 255) / 256;

  // degree -> dis
  init_deg_kernel<<<(N + 255) / 256, 256, 0, stream>>>(dis, N);
  count_deg_kernel<<<(unsigned)eb256, 256, 0, stream>>>(dst, dis, E);
  dis_kernel<<<(N + 255) / 256, 256, 0, stream>>>(dis, N);

  // ---- layer 1
  gemm1_wmma_kernel<<<mtiles, 128, 0, stream>>>(x, W1, bufA, N);
  {
    long long tot = (long long)N * 64;
    init_agg_kernel<<<(unsigned)((tot + 255) / 256), 256, 0, stream>>>(
        bufA, dis, b1, bufB, N, 64);
  }
  edge_agg64_kernel<<<(unsigned)eb8, 256, 0, stream>>>(src, dst, bufA, dis, bufB, E);

  // ---- layer 2 (ReLU fused into GEMM2's A load; bufA reused for xw2)
  gemm2_wmma_kernel<<<mtiles, 96, 0, stream>>>(bufB, W2, bufA, N);
  {
    long long tot = (long long)N * 40;
    init_agg_kernel<<<(unsigned)((tot + 255) / 256), 256, 0, stream>>>(
        bufA, dis, b2, bufB, N, 40);
  }
  edge_agg40_kernel<<<(unsigned)eb8, 256, 0, stream>>>(src, dst, bufA, dis, bufB, E);

  // ---- log_softmax -> d_out
  logsoftmax40_kernel<<<(N * 32 + 255) / 256, 256, 0, stream>>>(bufB, out, N);
}